// Model3_81655918232021
// MI455X (gfx1250) — compile-verified
//
#include <hip/hip_runtime.h>
#include <hip/hip_bf16.h>

// ---------------- constants ----------------
#define N_     8000        // total nodes
#define NP_    1000        // nodes per batch
#define B_     8
#define E_     80000       // edges
#define EPB_   10000       // edges per batch (batch-blocked in edge_index)
#define HC_    32
#define F_     1000
#define TRI_T  499500
#define HID_   128
#define EPS_   1e-5f
#define NB1_   1024        // blocks for the big streaming GEMM

typedef float v2f __attribute__((ext_vector_type(2)));
typedef float v8f __attribute__((ext_vector_type(8)));

// ---------------- graph preprocessing ----------------
__global__ void k_deg(const int* __restrict__ src, const int* __restrict__ dst,
                      float* degf, int* cnt) {
  int e = blockIdx.x * 256 + threadIdx.x;
  if (e >= E_) return;
  atomicAdd(&degf[src[e]], 1.0f);   // exact integer-valued fp adds: deterministic
  atomicAdd(&cnt[dst[e]], 1);
}

__global__ void k_dinv(const float* degf, float* dinv) {
  int n = blockIdx.x * 256 + threadIdx.x;
  if (n >= N_) return;
  float d = degf[n];
  dinv[n] = d > 0.0f ? rsqrtf(d) : 0.0f;
}

__global__ void k_w(const int* src, const int* dst, const float* dinv, float* wv) {
  int e = blockIdx.x * 256 + threadIdx.x;
  if (e >= E_) return;
  wv[e] = -dinv[src[e]] * dinv[dst[e]];
}

// exclusive prefix sum of cnt[0..N_-1] -> rowptr (single 1024-thread block)
__global__ __launch_bounds__(1024)
void k_scan(const int* cnt, int* rowptr) {
  __shared__ int ssum[1024];
  int tid = threadIdx.x;
  int base = tid * 8;
  int loc[8]; int s = 0;
#pragma unroll
  for (int i = 0; i < 8; i++) { int idx = base + i; int v = (idx < N_) ? cnt[idx] : 0; loc[i] = s; s += v; }
  ssum[tid] = s; __syncthreads();
  for (int off = 1; off < 1024; off <<= 1) {
    int v = (tid >= off) ? ssum[tid - off] : 0;
    __syncthreads();
    ssum[tid] += v;
    __syncthreads();
  }
  int carry = (tid > 0) ? ssum[tid - 1] : 0;
#pragma unroll
  for (int i = 0; i < 8; i++) { int idx = base + i; if (idx < N_) rowptr[idx] = carry + loc[i]; }
  if (tid == 0) rowptr[N_] = ssum[1023];
}

// deterministic CSR fill: one block per dst row; 256 threads chunk-scan the
// batch's edge slice, LDS prefix-scan of match counts preserves edge order.
__global__ void k_fill(const int* __restrict__ src, const int* __restrict__ dst,
                       const float* __restrict__ wv, const int* __restrict__ rowptr,
                       int* esrc, float* ew) {
  __shared__ int scnt[256];
  int r = blockIdx.x;
  int tid = threadIdx.x;
  int b = r / NP_;
  int e0 = b * EPB_;
  const int CH = (EPB_ + 255) / 256;          // 40
  int c0 = e0 + tid * CH;
  int c1 = min(e0 + EPB_, c0 + CH);
  int cl = 0;
  for (int e = c0; e < c1; e++) if (dst[e] == r) cl++;
  scnt[tid] = cl; __syncthreads();
  for (int off = 1; off < 256; off <<= 1) {
    int v = (tid >= off) ? scnt[tid - off] : 0;
    __syncthreads();
    scnt[tid] += v;
    __syncthreads();
  }
  int pos = rowptr[r] + (tid > 0 ? scnt[tid - 1] : 0);
  for (int e = c0; e < c1; e++) {
    if (dst[e] == r) { esrc[pos] = src[e]; ew[pos] = wv[e]; pos++; }
  }
}

// Tx1 = prop(xid): row r gets ew scattered to column (src % NP)
__global__ void k_scatter_tx1(const int* __restrict__ rowptr, const int* __restrict__ esrc,
                              const float* __restrict__ ew, float* out) {
  int r = blockIdx.x;
  float* row = out + (size_t)r * NP_;
  if (threadIdx.x < NP_ / 4) {
    float4 z = {0.f, 0.f, 0.f, 0.f};
    ((float4*)row)[threadIdx.x] = z;
  }
  __syncthreads();
  if (threadIdx.x == 0) {
    int e0 = rowptr[r], e1 = rowptr[r + 1];
    for (int e = e0; e < e1; e++) row[esrc[e] % NP_] += ew[e];
  }
}

// dense prop, width 1000 (float4-vectorized: 250 active lanes x 4 cols):
// out[r,:] = 2*sum_e w_e * in[src_e,:] - prev[r,:]   (identityPrev: prev == I)
__global__ void k_prop_dense(const float* __restrict__ in, const float* __restrict__ prev,
                             float* __restrict__ out, const int* __restrict__ rowptr,
                             const int* __restrict__ esrc, const float* __restrict__ ew,
                             int identityPrev) {
  __shared__ int   s_src[64];
  __shared__ float s_w[64];
  int r = blockIdx.x;
  int tid = threadIdx.x;
  int c0 = tid * 4;                      // 16B aligned (row stride 4000B is 16B mult)
  bool act = c0 < NP_;                   // 250 active threads
  int e0 = rowptr[r], e1 = rowptr[r + 1];
  float4 acc = {0.f, 0.f, 0.f, 0.f};
  for (int baseE = e0; baseE < e1; baseE += 64) {
    int m = min(64, e1 - baseE);
    if (tid < m) { s_src[tid] = esrc[baseE + tid]; s_w[tid] = ew[baseE + tid]; }
    __syncthreads();
    if (act) {
      for (int k = 0; k < m; k++) {
        const float4 v = *(const float4*)(in + (size_t)s_src[k] * NP_ + c0);
        float wk = s_w[k];
        acc.x += wk * v.x; acc.y += wk * v.y; acc.z += wk * v.z; acc.w += wk * v.w;
      }
    }
    __syncthreads();
  }
  if (act) {
    int rl = r % NP_;
    float4 p;
    if (identityPrev) {
      p.x = (c0 + 0 == rl) ? 1.f : 0.f; p.y = (c0 + 1 == rl) ? 1.f : 0.f;
      p.z = (c0 + 2 == rl) ? 1.f : 0.f; p.w = (c0 + 3 == rl) ? 1.f : 0.f;
    } else {
      p = *(const float4*)(prev + (size_t)r * NP_ + c0);
    }
    float4 o;
    o.x = 2.f * acc.x - p.x; o.y = 2.f * acc.y - p.y;
    o.z = 2.f * acc.z - p.z; o.w = 2.f * acc.w - p.w;
    *(float4*)(out + (size_t)r * NP_ + c0) = o;
  }
}

// prop on 32-wide features: out[r,c] = alpha*sum_e w_e*in[src_e,c] - prev[r,c]
__global__ void k_prop32(const float* __restrict__ in, const float* __restrict__ prev,
                         float* __restrict__ out, const int* __restrict__ rowptr,
                         const int* __restrict__ esrc, const float* __restrict__ ew,
                         float alpha) {
  int r = blockIdx.x * 8 + (threadIdx.x >> 5);
  int lane = threadIdx.x & 31;
  if (r >= N_) return;
  int e0 = rowptr[r], e1 = rowptr[r + 1];
  float acc = 0.f;
  for (int e = e0; e < e1; e++) acc += ew[e] * in[(size_t)esrc[e] * HC_ + lane];
  float p = prev ? prev[(size_t)r * HC_ + lane] : 0.0f;
  out[(size_t)r * HC_ + lane] = alpha * acc - p;
}

// ---------------- WMMA projection: Cout(8000x32) += A(8000xK) @ Bw(Kx32) ----
// one wave per 16x16 tile, f32 16x16x4 WMMA, two accumulator chains for ILP
__global__ __launch_bounds__(128)
void k_proj(const float* __restrict__ A, const float* __restrict__ Bw,
            float* __restrict__ Cout, int lda, int ksteps, int tiles) {
  int wave = (blockIdx.x * blockDim.x + threadIdx.x) >> 5;
  if (wave >= tiles) return;                     // wave-uniform: EXEC stays all-ones
  int lane = threadIdx.x & 31;
  int mt = wave >> 1, nt = wave & 1;
  int half = lane >> 4, l16 = lane & 15;
  int row = mt * 16 + l16;
  int col = nt * 16 + l16;
  const float* arow = A  + (size_t)row * lda + half * 2;
  const float* bcol = Bw + (size_t)(half * 2) * HC_ + col;
  v8f c0 = {}; v8f c1 = {};
  int ks = 0;
  for (; ks + 1 < ksteps; ks += 2) {
    v2f a0, b0, a1, b1;
    a0.x = arow[0]; a0.y = arow[1];
    b0.x = bcol[0]; b0.y = bcol[HC_];
    a1.x = arow[4]; a1.y = arow[5];
    b1.x = bcol[4 * HC_]; b1.y = bcol[5 * HC_];
    c0 = __builtin_amdgcn_wmma_f32_16x16x4_f32(false, a0, false, b0, (short)0, c0, false, false);
    c1 = __builtin_amdgcn_wmma_f32_16x16x4_f32(false, a1, false, b1, (short)0, c1, false, false);
    arow += 8; bcol += 8 * HC_;
  }
  for (; ks < ksteps; ks++) {
    v2f a, b;
    a.x = arow[0]; a.y = arow[1];
    b.x = bcol[0]; b.y = bcol[HC_];
    c0 = __builtin_amdgcn_wmma_f32_16x16x4_f32(false, a, false, b, (short)0, c0, false, false);
    arow += 4; bcol += 4 * HC_;
  }
  float* crow = Cout + (size_t)(mt * 16 + half * 8) * HC_ + col;
#pragma unroll
  for (int v = 0; v < 8; v++) crow[(size_t)v * HC_] += c0[v] + c1[v];
}

// layer-1 init: out = I@Ws0 + bias  (row r of Ws0 broadcast per batch)
__global__ void k_init_out1(const float* __restrict__ w0, const float* __restrict__ b0,
                            float* out) {
  int idx = blockIdx.x * 256 + threadIdx.x;          // 8000*32 exact
  int r = idx >> 5, h = idx & 31;
  out[idx] = w0[(size_t)(r % NP_) * HC_ + h] + b0[h];
}

__global__ void k_init_bias(const float* __restrict__ b, float* out) {
  int idx = blockIdx.x * 256 + threadIdx.x;
  out[idx] = b[idx & 31];
}

__global__ void k_tanh(float* x) {
  int idx = blockIdx.x * 256 + threadIdx.x;
  x[idx] = tanhf(x[idx]);
}

// ---------------- pooling (two-stage, deterministic order) + hidden BN ------
__global__ void k_pool1(const float* __restrict__ h1, const float* __restrict__ h2,
                        const float* __restrict__ h3, float* pp) {
  int b = blockIdx.x / 10, ch = blockIdx.x % 10;   // 80 blocks
  int j = threadIdx.x;                              // 0..95
  int which = j / HC_, c = j % HC_;
  const float* H = which == 0 ? h1 : (which == 1 ? h2 : h3);
  float s = 0.f;
  int r0 = ch * 100;
  for (int r = r0; r < r0 + 100; r++) s += H[(size_t)(b * NP_ + r) * HC_ + c];
  pp[(b * 10 + ch) * 96 + j] = s;
}

__global__ void k_pool2(const float* __restrict__ pp, float* emb) {
  int idx = threadIdx.x;                            // 0..767
  int b = idx / 96, j = idx % 96;
  float s = 0.f;
  for (int ch = 0; ch < 10; ch++) s += pp[(b * 10 + ch) * 96 + j];
  emb[b * 96 + j] = s / (float)NP_;
}

__global__ void k_hbn(const float* __restrict__ emb, const float* __restrict__ g,
                      const float* __restrict__ bb, float* outp) {
  int j = threadIdx.x;          // 0..95
  float m = 0.f;
  for (int b = 0; b < B_; b++) m += emb[b * 96 + j];
  m *= (1.0f / B_);
  float v = 0.f;
  for (int b = 0; b < B_; b++) { float d = emb[b * 96 + j] - m; v += d * d; }
  v *= (1.0f / B_);
  float rs = rsqrtf(v + EPS_);
  for (int b = 0; b < B_; b++)
    outp[b * 96 + j] = (emb[b * 96 + j] - m) * rs * g[j] + bb[j];
}

// ---------------- triu feature batchnorm (stored as fbn8[t*8+b]) ----------------
__global__ void k_fstats(const float* __restrict__ x, const float* __restrict__ g,
                         const float* __restrict__ bb, float* __restrict__ fbn8) {
  int i = blockIdx.x;                       // row 0..998
  int tbase = i * (2 * F_ - i - 1) / 2;
  for (int j = i + 1 + threadIdx.x; j < F_; j += 256) {
    int t = tbase + (j - i - 1);
    float v[8]; float m = 0.f;
#pragma unroll
    for (int b = 0; b < B_; b++) { v[b] = x[(size_t)b * F_ * F_ + (size_t)i * F_ + j]; m += v[b]; }
    m *= (1.0f / B_);
    float var = 0.f;
#pragma unroll
    for (int b = 0; b < B_; b++) { float d = v[b] - m; var += d * d; }
    var *= (1.0f / B_);
    float rs = rsqrtf(var + EPS_);
    float gg = g[t], bt = bb[t];
#pragma unroll
    for (int b = 0; b < B_; b++) fbn8[(size_t)t * 8 + b] = (v[b] - m) * rs * gg + bt;
  }
}

// ---------------- big streaming GEMM: a1(8x128) = fbn8 @ W1 ------------------
// 1024 blocks x 128 threads; 4 t-groups x 32 h-quads; float4 (b128) W1 stream.
__global__ __launch_bounds__(128)
void k_mlp1(const float* __restrict__ fbn8, const float* __restrict__ W1,
            float* __restrict__ partial, int nblocks) {
  __shared__ float sf[32 * 8];
  int per = (TRI_T + nblocks - 1) / nblocks;
  int t0 = blockIdx.x * per;
  int t1 = min(TRI_T, t0 + per);
  int hq = threadIdx.x & 31;          // h columns 4*hq .. 4*hq+3
  int tg = threadIdx.x >> 5;          // t-group 0..3
  float acc[8][4];
#pragma unroll
  for (int b = 0; b < 8; b++)
#pragma unroll
    for (int j = 0; j < 4; j++) acc[b][j] = 0.f;
  for (int tc = t0; tc < t1; tc += 32) {
    int m = min(32, t1 - tc);
    // stage m*8 pre-normalized features (float4 coalesced)
    if (threadIdx.x < m * 2)
      ((float4*)sf)[threadIdx.x] = ((const float4*)(fbn8 + (size_t)tc * 8))[threadIdx.x];
    __syncthreads();
    __builtin_prefetch(W1 + (size_t)(tc + 32) * HID_ + (size_t)threadIdx.x, 0, 1);
    for (int tl = tg; tl < m; tl += 4) {
      const float4 w = *(const float4*)(W1 + (size_t)(tc + tl) * HID_ + hq * 4);
#pragma unroll
      for (int b = 0; b < 8; b++) {
        float f = sf[tl * 8 + b];
        acc[b][0] += f * w.x; acc[b][1] += f * w.y;
        acc[b][2] += f * w.z; acc[b][3] += f * w.w;
      }
    }
    __syncthreads();
  }
  float* p = partial + ((size_t)blockIdx.x * 4 + tg) * 1024;
#pragma unroll
  for (int b = 0; b < 8; b++)
#pragma unroll
    for (int j = 0; j < 4; j++) p[b * HID_ + hq * 4 + j] = acc[b][j];
}

__global__ void k_mlp1_red(const float* __restrict__ partial, const float* __restrict__ b1,
                           float* a1, int nparts) {
  int o = blockIdx.x * 256 + threadIdx.x;   // 0..1023
  float s = 0.f;
  for (int k = 0; k < nparts; k++) s += partial[(size_t)k * 1024 + o];
  a1[o] = s + b1[o & (HID_ - 1)];
}

// ---------------- small tail ----------------
__global__ void k_bnrelu(const float* __restrict__ in, const float* __restrict__ g,
                         const float* __restrict__ bb, float* out, int C) {
  int c = threadIdx.x;
  if (c >= C) return;
  float m = 0.f;
  for (int b = 0; b < B_; b++) m += in[b * C + c];
  m *= (1.0f / B_);
  float v = 0.f;
  for (int b = 0; b < B_; b++) { float d = in[b * C + c] - m; v += d * d; }
  v *= (1.0f / B_);
  float rs = rsqrtf(v + EPS_);
  for (int b = 0; b < B_; b++) {
    float z = (in[b * C + c] - m) * rs * g[c] + bb[c];
    out[b * C + c] = z > 0.f ? z : 0.f;
  }
}

__global__ void k_gemm_small(const float* __restrict__ in, const float* __restrict__ W,
                             const float* __restrict__ bias, float* out, int K, int Nc) {
  int tid = blockIdx.x * blockDim.x + threadIdx.x;
  if (tid >= B_ * Nc) return;
  int b = tid / Nc, n = tid % Nc;
  float s = bias[n];
  for (int k = 0; k < K; k++) s += in[b * K + k] * W[k * Nc + n];
  out[b * Nc + n] = s;
}

__global__ void k_logsoftmax(const float* __restrict__ logits, float* outp) {
  int b = threadIdx.x;
  if (b >= B_) return;
  float z0 = logits[b * 2], z1 = logits[b * 2 + 1];
  float m = fmaxf(z0, z1);
  float lse = m + logf(expf(z0 - m) + expf(z1 - m));
  outp[b * 2]     = z0 - lse;
  outp[b * 2 + 1] = z1 - lse;
}

// ---------------- host launcher ----------------
extern "C" void kernel_launch(void* const* d_in, const int* in_sizes, int n_in,
                              void* d_out, int out_size, void* d_ws, size_t ws_size,
                              hipStream_t stream) {
  const float* x    = (const float*)d_in[0];
  const int*   src  = (const int*)d_in[1];
  const int*   dst  = ((const int*)d_in[1]) + E_;
  const float* cw0  = (const float*)d_in[3];
  const float* cb0  = (const float*)d_in[4];
  const float* cw1  = (const float*)d_in[5];
  const float* cb1  = (const float*)d_in[6];
  const float* cw2  = (const float*)d_in[7];
  const float* cb2  = (const float*)d_in[8];
  const float* bnhg = (const float*)d_in[9];
  const float* bnhb = (const float*)d_in[10];
  const float* bng  = (const float*)d_in[11];
  const float* bnb  = (const float*)d_in[12];
  const float* W1   = (const float*)d_in[13];
  const float* b1   = (const float*)d_in[14];
  const float* bn1g = (const float*)d_in[15];
  const float* bn1b = (const float*)d_in[16];
  const float* W2   = (const float*)d_in[17];
  const float* b2   = (const float*)d_in[18];
  const float* bn2g = (const float*)d_in[19];
  const float* bn2b = (const float*)d_in[20];
  const float* W3   = (const float*)d_in[21];
  const float* b3   = (const float*)d_in[22];
  const float* bn3g = (const float*)d_in[23];
  const float* bn3b = (const float*)d_in[24];
  const float* W4   = (const float*)d_in[25];
  const float* b4   = (const float*)d_in[26];
  float* outp = (float*)d_out;                 // [0,16): logp ; [16,784): hbn

  char* ws = (char*)d_ws;
  size_t off = 0;
  auto alloc = [&](size_t bytes) -> void* {
    void* p = ws + off;
    off = (off + bytes + 255) & ~(size_t)255;
    return p;
  };
  float* degf   = (float*)alloc(N_ * 4);
  int*   cnt    = (int*)  alloc(N_ * 4);
  float* dinv   = (float*)alloc(N_ * 4);
  float* wv     = (float*)alloc(E_ * 4);
  int*   rowptr = (int*)  alloc((N_ + 1) * 4);
  int*   esrc   = (int*)  alloc(E_ * 4);
  float* ew     = (float*)alloc(E_ * 4);
  float* TxA    = (float*)alloc((size_t)N_ * NP_ * 4);
  float* TxB    = (float*)alloc((size_t)N_ * NP_ * 4);
  float* TxC    = (float*)alloc((size_t)N_ * NP_ * 4);
  float* h1     = (float*)alloc((size_t)N_ * HC_ * 4);
  float* h2     = (float*)alloc((size_t)N_ * HC_ * 4);
  float* h3     = (float*)alloc((size_t)N_ * HC_ * 4);
  float* tzA    = (float*)alloc((size_t)N_ * HC_ * 4);
  float* tzB    = (float*)alloc((size_t)N_ * HC_ * 4);
  float* tzC    = (float*)alloc((size_t)N_ * HC_ * 4);
  float* emb    = (float*)alloc(B_ * 96 * 4);
  float* pp     = (float*)alloc(80 * 96 * 4);
  float* fbn8   = (float*)alloc((size_t)TRI_T * 8 * 4);
  float* partial= (float*)alloc((size_t)NB1_ * 4 * 1024 * 4);
  float* a1     = (float*)alloc(B_ * HID_ * 4);
  float* a1r    = (float*)alloc(B_ * HID_ * 4);
  float* a2     = (float*)alloc(B_ * 64 * 4);
  float* a2r    = (float*)alloc(B_ * 64 * 4);
  float* a3     = (float*)alloc(B_ * 64 * 4);
  float* a3r    = (float*)alloc(B_ * 64 * 4);
  float* logits = (float*)alloc(B_ * 2 * 4);

  hipMemsetAsync(degf, 0, N_ * 4, stream);
  hipMemsetAsync(cnt,  0, N_ * 4, stream);

  // graph prep
  k_deg <<<(E_ + 255) / 256, 256, 0, stream>>>(src, dst, degf, cnt);
  k_dinv<<<(N_ + 255) / 256, 256, 0, stream>>>(degf, dinv);
  k_w   <<<(E_ + 255) / 256, 256, 0, stream>>>(src, dst, dinv, wv);
  k_scan<<<1, 1024, 0, stream>>>(cnt, rowptr);
  k_fill<<<N_, 256, 0, stream>>>(src, dst, wv, rowptr, esrc, ew);

  // ---- Cheb layer 1 (identity input): dense Tx in L2, WMMA projections ----
  const int elems32 = N_ * HC_;                       // 256000
  k_scatter_tx1<<<N_, 256, 0, stream>>>(rowptr, esrc, ew, TxA);   // Tx1
  k_init_out1<<<elems32 / 256, 256, 0, stream>>>(cw0, cb0, h1);   // I@Ws0 + b
  const int tiles = (N_ / 16) * 2;                    // 1000
  const int pgrid = (tiles * 32 + 127) / 128;         // 250
  k_proj<<<pgrid, 128, 0, stream>>>(TxA, cw0 + 1 * F_ * HC_, h1, NP_, NP_ / 4, tiles);
  k_prop_dense<<<N_, 256, 0, stream>>>(TxA, nullptr, TxB, rowptr, esrc, ew, 1);   // Tx2
  k_proj<<<pgrid, 128, 0, stream>>>(TxB, cw0 + 2 * F_ * HC_, h1, NP_, NP_ / 4, tiles);
  k_prop_dense<<<N_, 256, 0, stream>>>(TxB, TxA, TxC, rowptr, esrc, ew, 0);       // Tx3
  k_proj<<<pgrid, 128, 0, stream>>>(TxC, cw0 + 3 * F_ * HC_, h1, NP_, NP_ / 4, tiles);
  k_prop_dense<<<N_, 256, 0, stream>>>(TxC, TxB, TxA, rowptr, esrc, ew, 0);       // Tx4
  k_proj<<<pgrid, 128, 0, stream>>>(TxA, cw0 + 4 * F_ * HC_, h1, NP_, NP_ / 4, tiles);
  k_tanh<<<elems32 / 256, 256, 0, stream>>>(h1);

  // ---- Cheb layers 2 & 3 (32-wide): sparse prop + WMMA projections ----
  const float* cws[2] = {cw1, cw2};
  const float* cbs[2] = {cb1, cb2};
  float* hin = h1;
  float* houts[2] = {h2, h3};
  for (int L = 0; L < 2; L++) {
    const float* cw = cws[L]; const float* cb = cbs[L]; float* ho = houts[L];
    k_init_bias<<<elems32 / 256, 256, 0, stream>>>(cb, ho);
    k_proj<<<pgrid, 128, 0, stream>>>(hin, cw + 0 * HC_ * HC_, ho, HC_, HC_ / 4, tiles);
    k_prop32<<<N_ / 8, 256, 0, stream>>>(hin, nullptr, tzA, rowptr, esrc, ew, 1.0f);  // Tx1
    k_proj<<<pgrid, 128, 0, stream>>>(tzA, cw + 1 * HC_ * HC_, ho, HC_, HC_ / 4, tiles);
    k_prop32<<<N_ / 8, 256, 0, stream>>>(tzA, hin, tzB, rowptr, esrc, ew, 2.0f);      // Tx2
    k_proj<<<pgrid, 128, 0, stream>>>(tzB, cw + 2 * HC_ * HC_, ho, HC_, HC_ / 4, tiles);
    k_prop32<<<N_ / 8, 256, 0, stream>>>(tzB, tzA, tzC, rowptr, esrc, ew, 2.0f);      // Tx3
    k_proj<<<pgrid, 128, 0, stream>>>(tzC, cw + 3 * HC_ * HC_, ho, HC_, HC_ / 4, tiles);
    k_prop32<<<N_ / 8, 256, 0, stream>>>(tzC, tzB, tzA, rowptr, esrc, ew, 2.0f);      // Tx4
    k_proj<<<pgrid, 128, 0, stream>>>(tzA, cw + 4 * HC_ * HC_, ho, HC_, HC_ / 4, tiles);
    k_tanh<<<elems32 / 256, 256, 0, stream>>>(ho);
    hin = ho;
  }

  // ---- pooling + hidden BN (second output) ----
  k_pool1<<<80, 96, 0, stream>>>(h1, h2, h3, pp);
  k_pool2<<<1, 768, 0, stream>>>(pp, emb);
  k_hbn <<<1, 96, 0, stream>>>(emb, bnhg, bnhb, outp + 16);

  // ---- feature BN + streaming 256MB GEMM + MLP tail ----
  k_fstats<<<F_ - 1, 256, 0, stream>>>(x, bng, bnb, fbn8);
  k_mlp1  <<<NB1_, 128, 0, stream>>>(fbn8, W1, partial, NB1_);
  k_mlp1_red<<<4, 256, 0, stream>>>(partial, b1, a1, NB1_ * 4);
  k_bnrelu<<<1, 128, 0, stream>>>(a1, bn1g, bn1b, a1r, HID_);
  k_gemm_small<<<1, 512, 0, stream>>>(a1r, W2, b2, a2, HID_, 64);
  k_bnrelu<<<1, 64, 0, stream>>>(a2, bn2g, bn2b, a2r, 64);
  k_gemm_small<<<1, 512, 0, stream>>>(a2r, W3, b3, a3, 64, 64);
  k_bnrelu<<<1, 64, 0, stream>>>(a3, bn3g, bn3b, a3r, 64);
  k_gemm_small<<<1, 16, 0, stream>>>(a3r, W4, b4, logits, 64, 2);
  k_logsoftmax<<<1, 8, 0, stream>>>(logits, outp);
  (void)in_sizes; (void)n_in; (void)out_size; (void)ws_size;
}